// MotifConv_25383256719491
// MI455X (gfx1250) — compile-verified
//
#include <hip/hip_runtime.h>
#include <math.h>

// Problem constants (match reference)
constexpr int    NN  = 50000;
constexpr int    C   = 128;
constexpr int    DD  = 64;
constexpr int    RR  = 4;
constexpr long long EE  = 400000;
constexpr long long EMM = 400000;

typedef float v2f __attribute__((ext_vector_type(2)));
typedef float v8f __attribute__((ext_vector_type(8)));

// D = A(16x4) * B(4x16) + C  — FP32 WMMA, wave32.
// A layout: lanes 0-15 hold M=lane, K=koff..koff+1 in 2 VGPRs (koff=0 lanes<16, 2 lanes>=16)
// B layout: lanes 0-15 hold N=lane, same K striping
// C/D:      N = lane&15, M = v + 8*(lane>=16) for VGPR v in 0..7
__device__ __forceinline__ v8f wmma4(v2f a, v2f b, v8f c) {
    return __builtin_amdgcn_wmma_f32_16x16x4_f32(
        /*neg_a=*/false, a, /*neg_b=*/false, b,
        /*c_mod=*/(short)0, c, /*reuse_a=*/false, /*reuse_b=*/false);
}

// gfx1250 async copy global -> LDS (ASYNCcnt-tracked), 16 bytes per lane.
__device__ __forceinline__ void async_ld_b128(unsigned lds_off, const void* gaddr) {
    asm volatile("global_load_async_to_lds_b128 %0, %1, off"
                 :: "v"(lds_off), "v"(gaddr)
                 : "memory");
}
__device__ __forceinline__ void wait_async0() {
    asm volatile("s_wait_asynccnt 0x0" ::: "memory");
}

// ---------------------------------------------------------------------------
// K1: per-relation scatter-add  agg[r][dst] += ew * x[src];  deg[dst] += 1
// ---------------------------------------------------------------------------
__global__ void rel_scatter_kernel(const float* __restrict__ x,
                                   const int* __restrict__ esrc,
                                   const int* __restrict__ edst,
                                   const float* __restrict__ ew,
                                   float* __restrict__ agg,
                                   float* __restrict__ deg) {
    const int lane = threadIdx.x & 31;
    long long eidx = (long long)blockIdx.x * 8 + (threadIdx.x >> 5);
    if (eidx >= RR * EE) return;
    const int r   = (int)(eidx / EE);
    const int src = esrc[eidx];
    const int dst = edst[eidx];
    const float w = ew[eidx];
    const float4 v = ((const float4*)(x + (size_t)src * C))[lane];
    float* ag = agg + ((size_t)r * NN + (size_t)dst) * C + lane * 4;
    atomicAdd(ag + 0, w * v.x);
    atomicAdd(ag + 1, w * v.y);
    atomicAdd(ag + 2, w * v.z);
    atomicAdd(ag + 3, w * v.w);
    if (lane == 0) atomicAdd(&deg[dst], 1.0f);
}

// ---------------------------------------------------------------------------
// K2: h = (sum_r agg_r @ W_r) * norm + x @ root + bias
// Block: 8 waves compute 16 rows x 128 cols; wave w owns col tile w*16.
// ---------------------------------------------------------------------------
__global__ void h_gemm_kernel(const float* __restrict__ agg,
                              const float* __restrict__ x,
                              const float* __restrict__ weight,
                              const float* __restrict__ root,
                              const float* __restrict__ bias,
                              const float* __restrict__ deg,
                              float* __restrict__ h) {
    const int lane = threadIdx.x & 31;
    const int wave = threadIdx.x >> 5;
    const int row0 = blockIdx.x * 16;
    const int m    = lane & 15;
    const int koff = (lane >> 4) << 1;          // 0 or 2
    const int col  = wave * 16 + m;

    v8f acc1 = {};   // message GEMM
    v8f acc2 = {};   // x @ root

    for (int r = 0; r < RR; ++r) {
        const float* arow = agg + ((size_t)r * NN + (size_t)(row0 + m)) * C;
        const float* B    = weight + (size_t)r * C * C;
#pragma unroll 4
        for (int k0 = 0; k0 < C; k0 += 4) {
            v2f a, b;
            a.x = arow[k0 + koff];
            a.y = arow[k0 + koff + 1];
            b.x = B[(k0 + koff) * C + col];
            b.y = B[(k0 + koff + 1) * C + col];
            acc1 = wmma4(a, b, acc1);
        }
    }
    {
        const float* arow = x + (size_t)(row0 + m) * C;
#pragma unroll 4
        for (int k0 = 0; k0 < C; k0 += 4) {
            v2f a, b;
            a.x = arow[k0 + koff];
            a.y = arow[k0 + koff + 1];
            b.x = root[(k0 + koff) * C + col];
            b.y = root[(k0 + koff + 1) * C + col];
            acc2 = wmma4(a, b, acc2);
        }
    }

    const float bcol    = bias[col];
    const int   rowBase = row0 + ((lane >> 4) << 3);
#pragma unroll
    for (int v = 0; v < 8; ++v) {
        const int row = rowBase + v;
        const float d = deg[row];
        const float nrm = d > 0.0f ? 1.0f / d : 0.0f;
        h[(size_t)row * C + col] = acc1[v] * nrm + acc2[v] + bcol;
    }
}

// ---------------------------------------------------------------------------
// K3: motif scatter  mspmm[mi][dst] += val * h[src]
// ---------------------------------------------------------------------------
__global__ void motif_scatter_kernel(const float* __restrict__ h,
                                     const int* __restrict__ msrc,
                                     const int* __restrict__ mdst,
                                     const float* __restrict__ mval,
                                     float* __restrict__ mspmm) {
    const int lane = threadIdx.x & 31;
    long long eidx = (long long)blockIdx.x * 8 + (threadIdx.x >> 5);
    if (eidx >= 13 * EMM) return;
    const int mi  = (int)(eidx / EMM);
    const int src = msrc[eidx];
    const int dst = mdst[eidx];
    const float w = mval[eidx];
    const float4 v = ((const float4*)(h + (size_t)src * C))[lane];
    float* dp = mspmm + ((size_t)mi * NN + (size_t)dst) * C + lane * 4;
    atomicAdd(dp + 0, w * v.x);
    atomicAdd(dp + 1, w * v.y);
    atomicAdd(dp + 2, w * v.z);
    atomicAdd(dp + 3, w * v.w);
}

// ---------------------------------------------------------------------------
// K4: per-motif attention fusion.
// grid = (N/16, 13). Block = 8 waves, 16 rows, motif i (=blockIdx.y+1).
// 14 K-blocks (b=0..12: c-components; b=13: mw) x 4 col-tiles.
// Waves split in 2 groups of 4; group g handles b = g, g+2, ..., staging its
// 16x128 A tile into LDS once via global_load_async_to_lds_b128 (ASYNCcnt),
// then 4 waves run the WMMA K-loop (A from LDS, B slice from global).
// Partial 16x16 tiles accumulate into lds_c / lds_mw via ds_add_f32.
// Finally att = sigmoid(rowdot(mw,c)); out = att * (mw - c).
// ---------------------------------------------------------------------------
__global__ void attention_kernel(const float* __restrict__ h,
                                 const float* __restrict__ mspmm,
                                 const float* __restrict__ wa,
                                 const float* __restrict__ ba,
                                 const float* __restrict__ motifW,
                                 const float* __restrict__ motifb,
                                 float* __restrict__ out) {
    const int i    = blockIdx.y + 1;     // motif index 1..13
    const int row0 = blockIdx.x * 16;

    __shared__ float ldsA[2][16 * 128];  // per-group staged A tile (8 KB each)
    __shared__ float lds_c[16 * 64];
    __shared__ float lds_mw[16 * 64];
    __shared__ float lds_att[16];

    for (int idx = threadIdx.x; idx < 16 * 64; idx += 256) {
        lds_c[idx]  = 0.0f;
        lds_mw[idx] = 0.0f;
    }
    __syncthreads();

    const int lane = threadIdx.x & 31;
    const int wave = threadIdx.x >> 5;
    const int g    = wave >> 2;                  // group 0/1
    const int ct   = wave & 3;                   // col tile within group
    const int lig  = ct * 32 + lane;             // lane id within group [0,128)
    const int m    = lane & 15;
    const int koff = (lane >> 4) << 1;
    const int col  = ct * 16 + m;

    for (int iter = 0; iter < 7; ++iter) {
        const int b = g + 2 * iter;              // K-block 0..13
        const float* A;
        const float* B;
        float* ldsT;
        if (b < 13) {
            const int j = (b < i) ? b : b + 1;   // tensor id (0=h, 1..13=mspmm)
            A = (j == 0) ? h : (mspmm + (size_t)(j - 1) * NN * C);
            B = motifW + ((size_t)(i - 1) * 13 * C + (size_t)b * C) * DD;
            ldsT = lds_c;
        } else {                                 // b == 13 -> mw block
            A = mspmm + (size_t)(i - 1) * NN * C;
            B = wa;
            ldsT = lds_mw;
        }

        // --- async-stage A tile (16 rows x 128 f32) into ldsA[g] ---
        // 512 x b128 chunks, 128 lanes -> 4 chunks per lane.
#pragma unroll
        for (int t = 0; t < 4; ++t) {
            const int idx = lig + 128 * t;       // chunk id [0,512)
            const int row = idx >> 5;            // 0..15
            const int cin = idx & 31;            // chunk within row
            const float* gsrc = A + (size_t)(row0 + row) * C + cin * 4;
            const unsigned loff =
                (unsigned)(size_t)(&ldsA[g][0]) + (unsigned)idx * 16u;
            async_ld_b128(loff, gsrc);
        }
        wait_async0();
        __syncthreads();

        // --- WMMA K-loop: A from LDS, B slice from global ---
        const float* arow = &ldsA[g][m * 128];
        v8f acc = {};
#pragma unroll 4
        for (int k0 = 0; k0 < C; k0 += 4) {
            v2f a = *(const v2f*)(arow + k0 + koff);   // ds_load_b64
            v2f bb;
            bb.x = B[(k0 + koff) * DD + col];
            bb.y = B[(k0 + koff + 1) * DD + col];
            acc = wmma4(a, bb, acc);
        }
        const int rbase = (lane >> 4) << 3;
#pragma unroll
        for (int v = 0; v < 8; ++v)
            atomicAdd(&ldsT[(rbase + v) * 64 + col], acc[v]);
        __syncthreads();                         // before re-staging ldsA
    }

    // add biases
    for (int idx = threadIdx.x; idx < 16 * 64; idx += 256) {
        const int cl = idx & 63;
        lds_c[idx]  += motifb[(i - 1) * DD + cl];
        lds_mw[idx] += ba[cl];
    }
    __syncthreads();

    if (threadIdx.x < 16) {
        float s = 0.0f;
#pragma unroll 8
        for (int cl = 0; cl < 64; ++cl)
            s += lds_mw[threadIdx.x * 64 + cl] * lds_c[threadIdx.x * 64 + cl];
        lds_att[threadIdx.x] = 1.0f / (1.0f + __expf(-s));
    }
    __syncthreads();

    for (int idx = threadIdx.x; idx < 16 * 64; idx += 256) {
        const int rr = idx >> 6;
        const int cl = idx & 63;
        out[(size_t)(row0 + rr) * (13 * DD) + (size_t)(i - 1) * DD + cl] =
            lds_att[rr] * (lds_mw[idx] - lds_c[idx]);
    }
}

// ---------------------------------------------------------------------------
extern "C" void kernel_launch(void* const* d_in, const int* in_sizes, int n_in,
                              void* d_out, int out_size, void* d_ws, size_t ws_size,
                              hipStream_t stream) {
    const float* x      = (const float*)d_in[0];
    const int*   esrc   = (const int*)  d_in[1];
    const int*   edst   = (const int*)  d_in[2];
    const float* ew     = (const float*)d_in[3];
    const int*   msrc   = (const int*)  d_in[4];
    const int*   mdst   = (const int*)  d_in[5];
    const float* mval   = (const float*)d_in[6];
    const float* weight = (const float*)d_in[7];
    const float* root   = (const float*)d_in[8];
    const float* bias   = (const float*)d_in[9];
    const float* wa     = (const float*)d_in[10];
    const float* ba     = (const float*)d_in[11];
    const float* motifW = (const float*)d_in[12];
    const float* motifb = (const float*)d_in[13];
    float* out = (float*)d_out;

    // Workspace layout: [region0: mspmm (13*N*C f32) aliased with agg (4*N*C f32)]
    //                   [h: N*C f32] [deg: N f32]
    const size_t mspmmBytes = 13ull * NN * C * sizeof(float);   // 332.8 MB
    const size_t aggBytes   = (size_t)RR * NN * C * sizeof(float);
    const size_t hBytes     = (size_t)NN * C * sizeof(float);
    const size_t degBytes   = (size_t)NN * sizeof(float);
    const size_t need       = mspmmBytes + hBytes + degBytes;
    if (ws_size < need) return;

    char*  ws      = (char*)d_ws;
    float* region0 = (float*)ws;                       // agg, then (re-zeroed) mspmm
    float* h       = (float*)(ws + mspmmBytes);
    float* deg     = (float*)(ws + mspmmBytes + hBytes);

    // Stage 1: relation scatter into agg
    hipMemsetAsync(region0, 0, aggBytes, stream);
    hipMemsetAsync(deg, 0, degBytes, stream);
    rel_scatter_kernel<<<(int)((RR * EE) / 8), 256, 0, stream>>>(x, esrc, edst, ew,
                                                                 region0, deg);
    // Stage 2: h GEMM (WMMA)
    h_gemm_kernel<<<NN / 16, 256, 0, stream>>>(region0, x, weight, root, bias, deg, h);

    // Stage 3: motif scatter (agg region reused as mspmm)
    hipMemsetAsync(region0, 0, mspmmBytes, stream);
    motif_scatter_kernel<<<(int)((13 * EMM) / 8), 256, 0, stream>>>(h, msrc, mdst, mval,
                                                                    region0);
    // Stage 4: attention fusion (WMMA + async-LDS staging + LDS reduce)
    dim3 g4(NN / 16, 13);
    attention_kernel<<<g4, 256, 0, stream>>>(h, region0, wa, ba, motifW, motifb, out);
}